// MotifClassifier_38809324486859
// MI455X (gfx1250) — compile-verified
//
#include <hip/hip_runtime.h>
#include <hip/hip_bf16.h>

#define Nn 8192
#define Ee 262144
#define Dd 256
#define Hh 256
#define Kk 3
#define Cc 2

typedef float v2f __attribute__((ext_vector_type(2)));
typedef float v8f __attribute__((ext_vector_type(8)));

// ---------------------------------------------------------------------------
// Degree counts (with self loops): deg_out[u]++ , deg_in[v]++ for E edges + N self loops
// ---------------------------------------------------------------------------
__global__ __launch_bounds__(256) void degree_kernel(const int* __restrict__ us,
                                                     const int* __restrict__ vs,
                                                     float* __restrict__ deg_out,
                                                     float* __restrict__ deg_in) {
  int e = blockIdx.x * blockDim.x + threadIdx.x;
  if (e >= Ee + Nn) return;
  int u = (e < Ee) ? us[e] : (e - Ee);
  int v = (e < Ee) ? vs[e] : (e - Ee);
  atomicAdd(&deg_out[u], 1.0f);
  atomicAdd(&deg_in[v], 1.0f);
}

// ---------------------------------------------------------------------------
// f32 WMMA GEMM: C[M x Nc] = A[M x Kd] @ B[Kd x Nc] (+ bias[Nc]) , all row-major.
// One wave per 16x16 output tile; 8 waves per block; K stepped by 4 via
// V_WMMA_F32_16X16X4_F32 (full fp32 precision, matches the f32 reference).
// A-frag layout (ISA 7.12.2, 32-bit A 16x4): lane = (half, l16),
//   VGPR j holds A[m0+l16][k + j + 2*half].
// B-frag (4x16, row striped across lanes): VGPR j holds B[k + j + 2*half][n0+l16].
// C/D: VGPR r, lane -> C[m0 + r + 8*half][n0 + l16].
// ---------------------------------------------------------------------------
__global__ __launch_bounds__(256) void wmma_gemm_f32(const float* __restrict__ A,
                                                     const float* __restrict__ B,
                                                     const float* __restrict__ bias,
                                                     float* __restrict__ C,
                                                     int M, int Kd, int Nc) {
  const int wave = threadIdx.x >> 5;          // 0..7
  const int lane = threadIdx.x & 31;
  const int half = lane >> 4;                 // 0 or 1
  const int l16  = lane & 15;
  const int n0 = blockIdx.x * 16;
  const int m0 = (blockIdx.y * 8 + wave) * 16;
  if (m0 >= M) return;                        // wave-uniform; EXEC stays all-1s

  v8f acc = {};
  const float* __restrict__ arow = A + (size_t)(m0 + l16) * Kd;
  for (int k = 0; k < Kd; k += 4) {
    v2f a, b;
    int ka = k + 2 * half;
    a.x = arow[ka + 0];
    a.y = arow[ka + 1];
    b.x = B[(size_t)(ka + 0) * Nc + n0 + l16];
    b.y = B[(size_t)(ka + 1) * Nc + n0 + l16];
    acc = __builtin_amdgcn_wmma_f32_16x16x4_f32(
        /*neg_a=*/false, a, /*neg_b=*/false, b,
        /*c_mod=*/(short)0, acc, /*reuse_a=*/false, /*reuse_b=*/false);
  }

  const float bv = bias ? bias[n0 + l16] : 0.0f;
#pragma unroll
  for (int r = 0; r < 8; ++r) {
    C[(size_t)(m0 + r + 8 * half) * Nc + n0 + l16] = acc[r] + bv;
  }
}

// ---------------------------------------------------------------------------
// conv scatter: conv[v] += hw[u] * rsqrt(deg_out[u]) for every edge/self-loop.
// One block per edge, one thread per feature (D=256). Rows are L2-resident.
// ---------------------------------------------------------------------------
__global__ __launch_bounds__(256) void scatter_conv(const int* __restrict__ us,
                                                    const int* __restrict__ vs,
                                                    const float* __restrict__ hw,
                                                    const float* __restrict__ deg_out,
                                                    float* __restrict__ conv) {
  int e = blockIdx.x;
  int d = threadIdx.x;
  int u = (e < Ee) ? us[e] : (e - Ee);
  int v = (e < Ee) ? vs[e] : (e - Ee);
  float s = rsqrtf(deg_out[u]);
  atomicAdd(&conv[(size_t)v * Dd + d], hw[(size_t)u * Dd + d] * s);
}

__global__ __launch_bounds__(256) void conv_finish(float* __restrict__ conv,
                                                   const float* __restrict__ deg_in,
                                                   const float* __restrict__ b_conv) {
  int i = blockIdx.x;
  int d = threadIdx.x;
  size_t idx = (size_t)i * Dd + d;
  conv[idx] = conv[idx] * rsqrtf(deg_in[i]) + b_conv[d];
}

// ---------------------------------------------------------------------------
// pools[0:H] = mean(Z, axis=0). 64 blocks x 128 rows, thread = column.
// pools pre-zeroed; contributions scaled by 1/N.
// ---------------------------------------------------------------------------
__global__ __launch_bounds__(256) void mean_pool(const float* __restrict__ Z,
                                                 float* __restrict__ pools0) {
  int col = threadIdx.x;
  int r0 = blockIdx.x * (Nn / 64);
  float acc = 0.0f;
  for (int n = r0; n < r0 + (Nn / 64); ++n) acc += Z[(size_t)n * Hh + col];
  atomicAdd(&pools0[col], acc * (1.0f / (float)Nn));
}

// ---------------------------------------------------------------------------
// Edge attention: t[e] = tanh(leaky_relu(dot(Z[u],wA[0:H]) + dot(Z[v],wA[H:2H]) + b))
// One wave per edge; lane-strided dot + shuffle reduction (wave32).
// ---------------------------------------------------------------------------
__global__ __launch_bounds__(256) void edge_attn(const float* __restrict__ Z,
                                                 const int* __restrict__ us,
                                                 const int* __restrict__ vs,
                                                 const float* __restrict__ w_attn,
                                                 const float* __restrict__ b_attn,
                                                 float* __restrict__ t_e) {
  int wave = threadIdx.x >> 5;
  int lane = threadIdx.x & 31;
  int e = blockIdx.x * 8 + wave;
  if (e >= Ee) return;
  int u = us[e], v = vs[e];
  float acc = 0.0f;
  const float* zu = Z + (size_t)u * Hh;
  const float* zv = Z + (size_t)v * Hh;
  for (int hh = lane; hh < Hh; hh += 32)
    acc += zu[hh] * w_attn[hh] + zv[hh] * w_attn[Hh + hh];
#pragma unroll
  for (int off = 16; off > 0; off >>= 1) acc += __shfl_down(acc, off, 32);
  if (lane == 0) {
    float x = acc + b_attn[0];
    float lr = (x > 0.0f) ? x : 0.01f * x;   // jax.nn.leaky_relu default slope
    t_e[e] = tanhf(lr);
  }
}

// ---------------------------------------------------------------------------
// Motif edge stats: for each edge, gather A_motifs[k,u,v] (random HBM gather),
// accumulate num/cnt at v and src_cnt at u when mask is set.
// ---------------------------------------------------------------------------
__global__ __launch_bounds__(256) void motif_edges(const float* __restrict__ A_motifs,
                                                   const int* __restrict__ us,
                                                   const int* __restrict__ vs,
                                                   const float* __restrict__ t_e,
                                                   float* __restrict__ num,
                                                   float* __restrict__ cnt,
                                                   float* __restrict__ scnt) {
  int e = blockIdx.x * blockDim.x + threadIdx.x;
  if (e >= Ee) return;
  int u = us[e], v = vs[e];
  float tv = t_e[e];
#pragma unroll
  for (int i = 0; i < Kk; ++i) {
    float a = A_motifs[(size_t)i * Nn * Nn + (size_t)u * Nn + v];
    if (a > 0.0f) {
      atomicAdd(&num[i * Nn + v], tv);
      atomicAdd(&cnt[i * Nn + v], 1.0f);
      atomicAdd(&scnt[i * Nn + u], 1.0f);
    }
  }
}

// ---------------------------------------------------------------------------
// Motif pooling, phase 1: accumulate masked h_k into pool_acc (zeroed) and
// count n_sub. 64 blocks x 128 rows; thread = column; col 0 adds n_sub once.
//   h_k[n] = cnt>0 ? az[n]*(num/max(cnt,1)) : Z[n], gated by insub = cnt+scnt>0.
// ---------------------------------------------------------------------------
__global__ __launch_bounds__(256) void motif_pool_acc(const float* __restrict__ Z,
                                                      const float* __restrict__ az,
                                                      const float* __restrict__ num,
                                                      const float* __restrict__ cnt,
                                                      const float* __restrict__ scnt,
                                                      float* __restrict__ pool_acc,
                                                      float* __restrict__ nsub) {
  int col = threadIdx.x;
  int r0 = blockIdx.x * (Nn / 64);
  float acc = 0.0f;
  float ns = 0.0f;
  for (int n = r0; n < r0 + (Nn / 64); ++n) {
    float c = cnt[n];
    if ((c + scnt[n]) > 0.0f) {
      ns += 1.0f;
      acc += (c > 0.0f) ? az[(size_t)n * Hh + col] * (num[n] / fmaxf(c, 1.0f))
                        : Z[(size_t)n * Hh + col];
    }
  }
  atomicAdd(&pool_acc[col], acc);
  if (col == 0) atomicAdd(nsub, ns);
}

__global__ __launch_bounds__(256) void motif_pool_fin(float* __restrict__ pool_acc,
                                                      const float* __restrict__ nsub) {
  int col = threadIdx.x;
  float ns = *nsub;
  pool_acc[col] = (ns > 0.0f) ? pool_acc[col] / ns : 0.0f;
}

// ---------------------------------------------------------------------------
// Final: out[c] = dot(pools[0:1024], w_lin[:,c]) + b_lin[c]
// ---------------------------------------------------------------------------
__global__ __launch_bounds__(256) void final_lin(const float* __restrict__ pools,
                                                 const float* __restrict__ w_lin,
                                                 const float* __restrict__ b_lin,
                                                 float* __restrict__ out) {
  __shared__ float red[256];
  int t = threadIdx.x;
  int c = blockIdx.x;
  float acc = 0.0f;
  for (int j = t; j < (Kk + 1) * Hh; j += 256) acc += pools[j] * w_lin[j * Cc + c];
  red[t] = acc;
  __syncthreads();
#pragma unroll
  for (int s = 128; s > 0; s >>= 1) {
    if (t < s) red[t] += red[t + s];
    __syncthreads();
  }
  if (t == 0) out[c] = red[0] + b_lin[c];
}

// ---------------------------------------------------------------------------
extern "C" void kernel_launch(void* const* d_in, const int* in_sizes, int n_in,
                              void* d_out, int out_size, void* d_ws, size_t ws_size,
                              hipStream_t stream) {
  const float* h        = (const float*)d_in[0];
  const float* A_motifs = (const float*)d_in[1];
  const float* w_conv   = (const float*)d_in[2];
  const float* b_conv   = (const float*)d_in[3];
  const float* w_W      = (const float*)d_in[4];
  const float* b_W      = (const float*)d_in[5];
  const float* w_attn   = (const float*)d_in[6];
  const float* b_attn   = (const float*)d_in[7];
  const float* w_k      = (const float*)d_in[8];
  const float* b_k      = (const float*)d_in[9];
  const float* w_lin    = (const float*)d_in[10];
  const float* b_lin    = (const float*)d_in[11];
  const int*   us       = (const int*)d_in[12];
  const int*   vs       = (const int*)d_in[13];
  float* out = (float*)d_out;

  // Workspace layout (floats)
  float* ws       = (float*)d_ws;
  float* deg_out  = ws;                         // N
  float* deg_in   = deg_out + Nn;               // N
  float* numA     = deg_in + Nn;                // K*N
  float* cntA     = numA + Kk * Nn;             // K*N
  float* scntA    = cntA + Kk * Nn;             // K*N
  float* nsubA    = scntA + Kk * Nn;            // K
  float* pools    = nsubA + Kk;                 // (K+1)*H
  float* t_e      = pools + (Kk + 1) * Hh;      // E
  float* hw       = t_e + Ee;                   // N*D
  float* conv     = hw + (size_t)Nn * Dd;       // N*D
  float* Zb       = conv + (size_t)Nn * Dd;     // N*H
  float* az       = Zb + (size_t)Nn * Hh;       // N*H

  // Zero accumulators: [deg_out .. pools] contiguous block, plus conv.
  size_t zero_elems = (size_t)(2 * Nn + 3 * Kk * Nn + Kk + (Kk + 1) * Hh);
  hipMemsetAsync(deg_out, 0, zero_elems * sizeof(float), stream);
  hipMemsetAsync(conv, 0, (size_t)Nn * Dd * sizeof(float), stream);

  // 1. degrees
  degree_kernel<<<(Ee + Nn + 255) / 256, 256, 0, stream>>>(us, vs, deg_out, deg_in);

  // 2. hw = h @ w_conv   (WMMA f32)
  wmma_gemm_f32<<<dim3(Dd / 16, Nn / 128), 256, 0, stream>>>(h, w_conv, nullptr, hw,
                                                             Nn, Dd, Dd);

  // 3. conv scatter + normalize + bias
  scatter_conv<<<Ee + Nn, 256, 0, stream>>>(us, vs, hw, deg_out, conv);
  conv_finish<<<Nn, 256, 0, stream>>>(conv, deg_in, b_conv);

  // 4. Z = conv @ w_W + b_W   (WMMA f32)
  wmma_gemm_f32<<<dim3(Hh / 16, Nn / 128), 256, 0, stream>>>(conv, w_W, b_W, Zb,
                                                             Nn, Dd, Hh);

  // 5. pools[0] = mean(Z)
  mean_pool<<<64, 256, 0, stream>>>(Zb, pools);

  // 6. edge attention scalar t
  edge_attn<<<Ee / 8, 256, 0, stream>>>(Zb, us, vs, w_attn, b_attn, t_e);

  // 7. motif edge stats (A_motifs random gather)
  motif_edges<<<Ee / 256, 256, 0, stream>>>(A_motifs, us, vs, t_e, numA, cntA, scntA);

  // 8. per-motif: az GEMM + masked pooling (stream-ordered; az buffer reused)
  for (int i = 0; i < Kk; ++i) {
    wmma_gemm_f32<<<dim3(Hh / 16, Nn / 128), 256, 0, stream>>>(
        Zb, w_k + (size_t)i * Hh * Hh, b_k + (size_t)i * Hh, az, Nn, Hh, Hh);
    motif_pool_acc<<<64, 256, 0, stream>>>(Zb, az, numA + i * Nn, cntA + i * Nn,
                                           scntA + i * Nn, pools + (i + 1) * Hh,
                                           nsubA + i);
    motif_pool_fin<<<1, 256, 0, stream>>>(pools + (i + 1) * Hh, nsubA + i);
  }

  // 9. final linear -> out[2]
  final_lin<<<Cc, 256, 0, stream>>>(pools, w_lin, b_lin, out);
}